// YoloLoss_62208306315984
// MI455X (gfx1250) — compile-verified
//
#include <hip/hip_runtime.h>
#include <hip/hip_bf16.h>
#include <math.h>

typedef __attribute__((ext_vector_type(2))) float v2f;
typedef __attribute__((ext_vector_type(8))) float v8f;

#define C_CLS 20
#define NCELL_B 49          // S*S = 49
#define OUT_CH 30           // C + 5*BOX
#define TGT_CH 25           // C + 5
#define S_F 7.0f
#define W_COORD 5.0f
#define W_SIZE 5.0f
#define W_NOOBJ 0.5f
#define EPSV 1e-6f

// Exact 32-lane sum using V_WMMA_F32_16X16X4_F32.
// A: 16x4 f32 striped so each (lane, A-VGPR) pair is a unique matrix slot.
// We place the partial in A-VGPR0, zero in A-VGPR1; B = all ones (4x16).
// D[m][n] = sum_k A[m][k]  =>  summing all 16 D rows == sum of all 32 partials.
// C/D layout: lanes 0-15 hold rows 0-7 (VGPR0..7), lanes 16-31
// hold rows 8-15, so lane-local sum of the 8 D regs + shfl_xor(16) finishes it.
__device__ __forceinline__ float wave_sum_wmma(float p) {
    v2f a; a.x = p;   a.y = 0.0f;
    v2f b; b.x = 1.0f; b.y = 1.0f;
    v8f c = {};
    c = __builtin_amdgcn_wmma_f32_16x16x4_f32(false, a, false, b,
                                              (short)0, c, false, false);
    float s = ((c[0] + c[1]) + (c[2] + c[3])) + ((c[4] + c[5]) + (c[6] + c[7]));
    s += __shfl_xor(s, 16, 32);
    return s;  // full wave sum in every lane
}

__device__ __forceinline__ float iou_xywh(float ax, float ay, float aw, float ah,
                                          float bx, float by, float bw, float bh) {
    float ax1 = ax - aw * 0.5f, ay1 = ay - ah * 0.5f;
    float ax2 = ax + aw * 0.5f, ay2 = ay + ah * 0.5f;
    float bx1 = bx - bw * 0.5f, by1 = by - bh * 0.5f;
    float bx2 = bx + bw * 0.5f, by2 = by + bh * 0.5f;
    float iw = fmaxf(fminf(ax2, bx2) - fmaxf(ax1, bx1), 0.0f);
    float ih = fmaxf(fminf(ay2, by2) - fmaxf(ay1, by1), 0.0f);
    float inter  = iw * ih;
    float area_a = fabsf((ax2 - ax1) * (ay2 - ay1));
    float area_b = fabsf((bx2 - bx1) * (by2 - by1));
    return inter / (area_a + area_b - inter + EPSV);
}

__device__ __forceinline__ float sign_sqrt(float v) {
    float sgn = (v > 0.0f) ? 1.0f : ((v < 0.0f) ? -1.0f : 0.0f);
    return sgn * sqrtf(fabsf(v) + EPSV);
}

__global__ void yolo_loss_partial(const float* __restrict__ out,
                                  const float* __restrict__ tgt,
                                  const float* __restrict__ cw,
                                  float* __restrict__ blocksum,
                                  int ncell) {
    const int i = blockIdx.x * blockDim.x + threadIdx.x;
    float partial = 0.0f;

    if (i < ncell) {
        const int bi = i / NCELL_B;
        const int sp = i - bi * NCELL_B;
        const float* o = out + (size_t)bi * (OUT_CH * NCELL_B) + sp; // ch stride 49
        const float* t = tgt + (size_t)bi * (TGT_CH * NCELL_B) + sp;

        // ---- class logits + log-softmax ----
        float ocls[C_CLS];
        float omax = -3.402823466e38f;
#pragma unroll
        for (int c = 0; c < C_CLS; ++c) {
            ocls[c] = o[c * NCELL_B];
            omax = fmaxf(omax, ocls[c]);
        }
        float se = 0.0f;
#pragma unroll
        for (int c = 0; c < C_CLS; ++c) se += __expf(ocls[c] - omax);
        const float lse = omax + __logf(se);

        const float tconf = t[C_CLS * NCELL_B];        // obj in {0,1}
        const float m = (tconf == 1.0f) ? 1.0f : 0.0f;

        float lclass = 0.0f;
#pragma unroll
        for (int c = 0; c < C_CLS; ++c)
            lclass += cw[c] * t[c * NCELL_B] * (ocls[c] - lse);
        lclass = -m * lclass;

        // ---- boxes: [conf, x, y, w, h] x 2 ----
        float b0[5], b1[5];
#pragma unroll
        for (int k = 0; k < 5; ++k) {
            b0[k] = o[(C_CLS + k) * NCELL_B];
            b1[k] = o[(C_CLS + 5 + k) * NCELL_B];
        }
        float tb[4];
#pragma unroll
        for (int k = 0; k < 4; ++k) tb[k] = t[(C_CLS + 1 + k) * NCELL_B];

        // IoU with scale [1,1,S,S]
        const float tx = tb[0], ty = tb[1], tw = tb[2] * S_F, th = tb[3] * S_F;
        const float iou0 = iou_xywh(b0[1], b0[2], b0[3] * S_F, b0[4] * S_F, tx, ty, tw, th);
        const float iou1 = iou_xywh(b1[1], b1[2], b1[3] * S_F, b1[4] * S_F, tx, ty, tw, th);
        const bool pick1 = (iou1 > iou0);              // argmax: first index wins ties

        const float pconf = pick1 ? b1[0] : b0[0];
        const float px = pick1 ? b1[1] : b0[1];
        const float py = pick1 ? b1[2] : b0[2];
        const float pw = sign_sqrt(pick1 ? b1[3] : b0[3]);
        const float ph = sign_sqrt(pick1 ? b1[4] : b0[4]);

        const float txy0 = tb[0], txy1 = tb[1];
        const float twr = sqrtf(t[(C_CLS + 3) * NCELL_B] + EPSV);
        const float thr = sqrtf(t[(C_CLS + 4) * NCELL_B] + EPSV);

        const float dx = px - txy0, dy = py - txy1;
        const float dw = pw - twr,  dh = ph - thr;
        const float dc = pconf - tconf;
        const float d0 = b0[0] - tconf, d1 = b1[0] - tconf;

        partial = W_COORD * m * (dx * dx + dy * dy)
                + W_SIZE  * m * (dw * dw + dh * dh)
                + m * dc * dc
                + W_NOOBJ * (1.0f - m) * (d0 * d0 + d1 * d1)
                + lclass;
    }

    // ---- reduction: WMMA wave sum -> LDS block sum ----
    const float wsum = wave_sum_wmma(partial);
    __shared__ float lsum[8];
    const int lane = threadIdx.x & 31;
    const int wid  = threadIdx.x >> 5;
    if (lane == 0) lsum[wid] = wsum;
    __syncthreads();
    if (threadIdx.x == 0) {
        float bs = 0.0f;
#pragma unroll
        for (int w = 0; w < 8; ++w) bs += lsum[w];
        blocksum[blockIdx.x] = bs;
    }
}

__global__ void yolo_loss_final(const float* __restrict__ blocksum, int nblocks,
                                float* __restrict__ d_out, float inv_batch) {
    float p = 0.0f;
    for (int i = threadIdx.x; i < nblocks; i += blockDim.x) p += blocksum[i];
    const float wsum = wave_sum_wmma(p);
    __shared__ float lsum[8];
    const int lane = threadIdx.x & 31;
    const int wid  = threadIdx.x >> 5;
    if (lane == 0) lsum[wid] = wsum;
    __syncthreads();
    if (threadIdx.x == 0) {
        float ts = 0.0f;
#pragma unroll
        for (int w = 0; w < 8; ++w) ts += lsum[w];
        d_out[0] = ts * inv_batch;
    }
}

extern "C" void kernel_launch(void* const* d_in, const int* in_sizes, int n_in,
                              void* d_out, int out_size, void* d_ws, size_t ws_size,
                              hipStream_t stream) {
    const float* output = (const float*)d_in[0];
    const float* target = (const float*)d_in[1];
    const float* cw     = (const float*)d_in[2];

    const int ncell   = in_sizes[0] / OUT_CH;      // batch * 49
    const int batch   = ncell / NCELL_B;
    const int nblocks = (ncell + 255) / 256;       // 1568 for batch=8192
    float* bsum = (float*)d_ws;

    yolo_loss_partial<<<nblocks, 256, 0, stream>>>(output, target, cw, bsum, ncell);
    yolo_loss_final<<<1, 256, 0, stream>>>(bsum, nblocks, (float*)d_out,
                                           1.0f / (float)batch);
}